// XEyTransformerLayer_377957122100
// MI455X (gfx1250) — compile-verified
//
#include <hip/hip_runtime.h>
#include <math.h>

typedef float v2f __attribute__((ext_vector_type(2)));
typedef float v8f __attribute__((ext_vector_type(8)));

#define NB 4
#define NN 256
#define DX 256
#define DE 64
#define DY 64
#define DFFX 1024
#define DFFE 256
#define LN_EPS 1e-5f

// ---- CDNA5 async LDS<->global data movement (ASYNCcnt-tracked, ISA ch.8) ----
// LDS operand = low 32 bits of generic shared address (ISA 10.2).
__device__ __forceinline__ unsigned lds_off(const void* p) {
  return (unsigned)(unsigned long long)p;
}
__device__ __forceinline__ void async_load_b128(unsigned dst_lds, const float* g) {
  asm volatile("global_load_async_to_lds_b128 %0, %1, off"
               :: "v"(dst_lds), "v"(g) : "memory");
}
__device__ __forceinline__ void async_store_b128(const float* g, unsigned src_lds) {
  asm volatile("global_store_async_from_lds_b128 %0, %1, off"
               :: "v"(g), "v"(src_lds) : "memory");
}
__device__ __forceinline__ void wait_async0() {
  asm volatile("s_wait_asynccnt 0x0" ::: "memory");
}

// ---- WMMA helper: C += A(16xK, LDS, row-major lda) @ B(KxN16, global, ldb) ----
// Fragment layouts per CDNA5 ISA 7.12.2 (f32 16x16x4):
//  A: lanes 0-15 -> rows 0-15, VGPR0=K0,VGPR1=K1 ; lanes 16-31 -> K2,K3
//  B: VGPR0 = row K0 (lanes 0-15) / K2 (lanes 16-31); VGPR1 = K1 / K3
//  D: VGPR v -> row (v + 8*half), lane&15 -> col
__device__ __forceinline__ v8f wmma_k(const float* A, int lda,
                                      const float* B, int ldb,
                                      int K, int lane, v8f acc) {
  const int half = lane >> 4;
  const int r = lane & 15;
  for (int k = 0; k < K; k += 4) {
    v2f a, b;
    a.x = A[r * lda + k + 2 * half];
    a.y = A[r * lda + k + 2 * half + 1];
    b.x = B[(k + 2 * half) * ldb + r];
    b.y = B[(k + 2 * half + 1) * ldb + r];
    acc = __builtin_amdgcn_wmma_f32_16x16x4_f32(false, a, false, b, (short)0,
                                                acc, false, false);
  }
  return acc;
}

// ---- LayerNorm over 16 LDS rows of width W (W divisible by 16), in-place ----
__device__ __forceinline__ void ln_rows16(float* S, int lds_, int W,
                                          const float* g, const float* be,
                                          float* red, int tid) {
  const int row = tid & 15;
  const int sub = tid >> 4;
  const int per = W >> 4;
  float s1 = 0.f, s2 = 0.f;
  for (int q = 0; q < per; ++q) {
    float v = S[row * lds_ + sub * per + q];
    s1 += v; s2 += v * v;
  }
  red[row * 16 + sub] = s1;
  red[256 + row * 16 + sub] = s2;
  __syncthreads();
  float t1 = 0.f, t2 = 0.f;
  for (int q = 0; q < 16; ++q) { t1 += red[row * 16 + q]; t2 += red[256 + row * 16 + q]; }
  float mean = t1 / (float)W;
  float inv = rsqrtf(t2 / (float)W - mean * mean + LN_EPS);
  for (int q = 0; q < per; ++q) {
    int cc = sub * per + q;
    float v = S[row * lds_ + cc];
    S[row * lds_ + cc] = (v - mean) * inv * g[cc] + be[cc];
  }
  __syncthreads();
}

// ---- per-batch y projections: ye_add, ye_mul, yx_add, yx_mul (each 256) ----
__global__ void k_yvec(const float* __restrict__ y,
                       const float* Wyea, const float* byea,
                       const float* Wyem, const float* byem,
                       const float* Wyxa, const float* byxa,
                       const float* Wyxm, const float* byxm,
                       float* yv) {
  int b = blockIdx.x, c = threadIdx.x;
  __shared__ float sy[DY];
  if (c < DY) sy[c] = y[b * DY + c];
  __syncthreads();
  float a0 = byea[c], a1 = byem[c], a2 = byxa[c], a3 = byxm[c];
  for (int k = 0; k < DY; ++k) {
    float yk = sy[k];
    a0 += yk * Wyea[k * DX + c];
    a1 += yk * Wyem[k * DX + c];
    a2 += yk * Wyxa[k * DX + c];
    a3 += yk * Wyxm[k * DX + c];
  }
  yv[b * 1024 + c] = a0;
  yv[b * 1024 + 256 + c] = a1;
  yv[b * 1024 + 512 + c] = a2;
  yv[b * 1024 + 768 + c] = a3;
}

// ---- QKV projections via WMMA; Q prescaled by 1/sqrt(df), all masked ----
__global__ void __launch_bounds__(256) k_qkv(
    const float* __restrict__ X, const unsigned char* __restrict__ mask,
    const float* Wq, const float* bq, const float* Wk, const float* bk,
    const float* Wv, const float* bv, float* Q, float* K, float* V) {
  const int mtile = blockIdx.x / 6;
  const int sect = blockIdx.x % 6;
  const int tid = threadIdx.x;
  const int wave = tid >> 5, lane = tid & 31;
  __shared__ float sX[16 * 260];
  __shared__ float sxm[16];
  // async-stage the X tile into LDS (16 rows x 256 = 1024 b128 chunks)
  for (int idx = tid; idx < 16 * 64; idx += 256) {
    int rr = idx >> 6, c4 = (idx & 63) << 2;
    async_load_b128(lds_off(&sX[rr * 260 + c4]), &X[(mtile * 16 + rr) * DX + c4]);
  }
  if (tid < 16) sxm[tid] = mask[mtile * 16 + tid] ? 1.f : 0.f;
  wait_async0();
  __syncthreads();
  int g = sect * 8 + wave;  // 0..47 = 3 proj x 16 ntiles
  int proj = g >> 4;
  int nt = g & 15;
  const float* W = proj == 0 ? Wq : (proj == 1 ? Wk : Wv);
  const float* bb = proj == 0 ? bq : (proj == 1 ? bk : bv);
  float* O = proj == 0 ? Q : (proj == 1 ? K : V);
  const float scale = (proj == 0) ? 0.17677669529663687f : 1.f;  // 1/sqrt(32)
  v8f acc = {};
  acc = wmma_k(sX, 260, W + nt * 16, DX, DX, lane, acc);
  const int half = lane >> 4, r = lane & 15;
#pragma unroll
  for (int v = 0; v < 8; ++v) {
    int m = v + 8 * half;
    int cc = nt * 16 + r;
    O[(mtile * 16 + m) * DX + cc] = (acc[v] + bb[cc]) * sxm[m] * scale;
  }
}

// ---- stats(X, axis=1): mean/min/max/std(ddof=1) -> (4, 1024) ----
__global__ void k_statsX(const float* __restrict__ X, float* sx) {
  int b = blockIdx.x, c = threadIdx.x;
  float s = 0.f, s2 = 0.f, mn = 1e30f, mx = -1e30f;
  for (int i = 0; i < NN; ++i) {
    float v = X[(b * NN + i) * DX + c];
    s += v; s2 += v * v; mn = fminf(mn, v); mx = fmaxf(mx, v);
  }
  float mean = s / (float)NN;
  float var = (s2 - s * s / (float)NN) / (float)(NN - 1);
  sx[b * 1024 + c] = mean;
  sx[b * 1024 + 256 + c] = mn;
  sx[b * 1024 + 512 + c] = mx;
  sx[b * 1024 + 768 + c] = sqrtf(fmaxf(var, 0.f));
}

// ---- stats(E, axes=(1,2)) -> (4, 256) ----
__global__ void k_statsE(const float* __restrict__ E, float* se) {
  int b = blockIdx.x, tid = threadIdx.x;
  int k = tid & 63, part = tid >> 6;  // 4 row-groups, 64 channels
  __shared__ float rs[256], rs2[256], rmn[256], rmx[256];
  const int M = NN * NN;
  float s = 0.f, s2 = 0.f, mn = 1e30f, mx = -1e30f;
  for (int e = part * (M / 4); e < (part + 1) * (M / 4); ++e) {
    float v = E[((size_t)b * M + e) * DE + k];
    s += v; s2 += v * v; mn = fminf(mn, v); mx = fmaxf(mx, v);
  }
  rs[tid] = s; rs2[tid] = s2; rmn[tid] = mn; rmx[tid] = mx;
  __syncthreads();
  if (tid < 64) {
    for (int p = 1; p < 4; ++p) {
      s += rs[p * 64 + tid]; s2 += rs2[p * 64 + tid];
      mn = fminf(mn, rmn[p * 64 + tid]); mx = fmaxf(mx, rmx[p * 64 + tid]);
    }
    float mean = s / (float)M;
    float var = (s2 - s * s / (float)M) / (float)(M - 1);
    se[b * 256 + tid] = mean;
    se[b * 256 + 64 + tid] = mn;
    se[b * 256 + 128 + tid] = mx;
    se[b * 256 + 192 + tid] = sqrtf(fmaxf(var, 0.f));
  }
}

// ---- fused edge/attention kernel: one block per (b,i) row ----
__global__ void __launch_bounds__(256) k_edge(
    const float* __restrict__ Ein, const unsigned char* __restrict__ mask,
    const float* __restrict__ Q, const float* __restrict__ K,
    const float* __restrict__ V, const float* __restrict__ yv,
    const float* Wem, const float* bem,    // e_mul (64x256)
    const float* Wea, const float* bea,    // e_add (64x256)
    const float* Weo, const float* beo,    // e_out (256x64)
    const float* gE1, const float* bE1,    // normE1
    const float* W1, const float* b1,      // linE1 (64x256)
    const float* W2, const float* b2,      // linE2 (256x64)
    const float* gE2, const float* bE2,    // normE2
    float* outE, float* wV) {
  const int b = blockIdx.x >> 8;
  const int i = blockIdx.x & 255;
  const int tid = threadIdx.x;
  const int wave = tid >> 5, lane = tid & 31;
  const int half = lane >> 4, r = lane & 15;

  __shared__ float sK[16 * 260];  // K tile, then V tile
  __shared__ float sA[16 * 260];  // Ytot, then FFN hidden
  __shared__ float sB[16 * 260];  // pre-e_out buffer
  __shared__ float sE[16 * 68];   // raw E tile, then FFN2 output
  __shared__ float sN[16 * 68];   // e_out result / LN / residual
  __shared__ float sQ[256];
  __shared__ float syeA[256];
  __shared__ float syeM[256];
  __shared__ float sxm[16];
  __shared__ float red[512];

  const float xmi = mask[b * NN + i] ? 1.f : 0.f;
  sQ[tid] = Q[(b * NN + i) * DX + tid];  // prescaled by 1/sqrt(df), masked
  syeA[tid] = yv[b * 1024 + tid];        // ye_add
  syeM[tid] = yv[b * 1024 + 256 + tid];  // ye_mul

  float om = -INFINITY, os = 0.f, ow = 0.f;  // online softmax state, channel=tid

  for (int jt = 0; jt < 16; ++jt) {
    __syncthreads();
    // async-stage K tile (16x256) and E tile (16x64) into LDS
    for (int idx = tid; idx < 16 * 64; idx += 256) {
      int rr = idx >> 6, c4 = (idx & 63) << 2;
      async_load_b128(lds_off(&sK[rr * 260 + c4]),
                      &K[(b * NN + jt * 16 + rr) * DX + c4]);
    }
    {
      int rr = tid >> 4, c4 = (tid & 15) << 2;
      async_load_b128(lds_off(&sE[rr * 68 + c4]),
                      &Ein[(((size_t)b * NN + i) * NN + jt * 16 + rr) * DE + c4]);
    }
    if (tid < 16) sxm[tid] = mask[b * NN + jt * 16 + tid] ? 1.f : 0.f;
    wait_async0();  // also drains last iteration's async E' stores before sN reuse
    __syncthreads();
    // E1/E2 projections fused into Ytot (sA) and ye-transformed pre (sB)
    for (int t = 0; t < 2; ++t) {
      int nt = wave * 2 + t;
      v8f a1 = {}, a2 = {};
      a1 = wmma_k(sE, 68, Wem + nt * 16, DX, DE, lane, a1);
      a2 = wmma_k(sE, 68, Wea + nt * 16, DX, DE, lane, a2);
#pragma unroll
      for (int v = 0; v < 8; ++v) {
        int m = v + 8 * half;
        int cc = nt * 16 + r;
        float ep = xmi * sxm[m];
        float e1 = (a1[v] + bem[cc]) * ep;
        float e2 = (a2[v] + bea[cc]) * ep;
        float yy = sQ[cc] * sK[m * 260 + cc] * (e1 + 1.f) + e2;
        sA[m * 260 + cc] = yy;
        sB[m * 260 + cc] = syeA[cc] + (syeM[cc] + 1.f) * yy;
      }
    }
    __syncthreads();
    // async-reload V tile over the K tile
    for (int idx = tid; idx < 16 * 64; idx += 256) {
      int rr = idx >> 6, c4 = (idx & 63) << 2;
      async_load_b128(lds_off(&sK[rr * 260 + c4]),
                      &V[(b * NN + jt * 16 + rr) * DX + c4]);
    }
    wait_async0();
    __syncthreads();
    // online softmax accumulation over 16 j's for this thread's channel
    for (int rr = 0; rr < 16; ++rr) {
      if (sxm[rr] != 0.f) {
        float yy = sA[rr * 260 + tid];
        float mn = fmaxf(om, yy);
        float ea = __expf(om - mn), eb = __expf(yy - mn);
        os = os * ea + eb;
        ow = ow * ea + eb * sK[rr * 260 + tid];
        om = mn;
      }
    }
    // e_out: 4 output tiles of 16x16, K=256 (waves 0..3)
    if (wave < 4) {
      v8f acc = {};
      acc = wmma_k(sB, 260, Weo + wave * 16, DE, DX, lane, acc);
#pragma unroll
      for (int v = 0; v < 8; ++v) {
        int m = v + 8 * half, cc = wave * 16 + r;
        sN[m * 68 + cc] = acc[v] + beo[cc];
      }
    }
    __syncthreads();
    // residual + LN1
    for (int idx = tid; idx < 16 * 64; idx += 256) {
      int rr = idx >> 6, cc = idx & 63;
      sN[rr * 68 + cc] = sE[rr * 68 + cc] + sN[rr * 68 + cc] * (xmi * sxm[rr]);
    }
    __syncthreads();
    ln_rows16(sN, 68, DE, gE1, bE1, red, tid);
    // FFN1: relu(sN @ W1 + b1) -> sA
    for (int t = 0; t < 2; ++t) {
      int nt = wave * 2 + t;
      v8f acc = {};
      acc = wmma_k(sN, 68, W1 + nt * 16, DFFE, DE, lane, acc);
#pragma unroll
      for (int v = 0; v < 8; ++v) {
        int m = v + 8 * half, cc = nt * 16 + r;
        sA[m * 260 + cc] = fmaxf(acc[v] + b1[cc], 0.f);
      }
    }
    __syncthreads();
    // FFN2: sA @ W2 + b2 -> sE (waves 0..3)
    if (wave < 4) {
      v8f acc = {};
      acc = wmma_k(sA, 260, W2 + wave * 16, DE, DFFE, lane, acc);
#pragma unroll
      for (int v = 0; v < 8; ++v) {
        int m = v + 8 * half, cc = wave * 16 + r;
        sE[m * 68 + cc] = acc[v] + b2[cc];
      }
    }
    __syncthreads();
    // residual2 + LN2 + async store E' tile from LDS
    for (int idx = tid; idx < 16 * 64; idx += 256) {
      int rr = idx >> 6, cc = idx & 63;
      sN[rr * 68 + cc] = sN[rr * 68 + cc] + sE[rr * 68 + cc];
    }
    __syncthreads();
    ln_rows16(sN, 68, DE, gE2, bE2, red, tid);
    {
      int rr = tid >> 4, c4 = (tid & 15) << 2;
      async_store_b128(&outE[(((size_t)b * NN + i) * NN + jt * 16 + rr) * DE + c4],
                       lds_off(&sN[rr * 68 + c4]));
    }
    // next-iteration wait_async0 drains these stores before sN/sE are reused;
    // S_ENDPGM performs an implicit wait-idle on the final iteration.
  }
  wV[(b * NN + i) * DX + tid] = (os > 0.f) ? ow / os : 0.f;
}

// ---- X path: x_out + LN1 + FFN(256->1024->256, chunked) + LN2 ----
__global__ void __launch_bounds__(256) k_xfinal(
    const float* __restrict__ X, const unsigned char* __restrict__ mask,
    const float* __restrict__ wV, const float* __restrict__ yv,
    const float* Wxo, const float* bxo, const float* gX1, const float* bX1,
    const float* W1, const float* b1,   // linX1 (256x1024)
    const float* W2, const float* b2,   // linX2 (1024x256)
    const float* gX2, const float* bX2, float* outX) {
  const int mtile = blockIdx.x;
  const int tid = threadIdx.x;
  const int wave = tid >> 5, lane = tid & 31;
  const int half = lane >> 4, r = lane & 15;
  __shared__ float sT[16 * 260];   // input t, reused for final output
  __shared__ float sX1[16 * 260];  // normed residual state
  __shared__ float sH[16 * 260];   // hidden chunk
  __shared__ float sxm[16];
  __shared__ float red[512];

  for (int idx = tid; idx < 16 * 256; idx += 256) {
    int rr = idx >> 8, cc = idx & 255;
    int row = mtile * 16 + rr;
    int b = row >> 8;
    sT[rr * 260 + cc] =
        yv[b * 1024 + 512 + cc] + (yv[b * 1024 + 768 + cc] + 1.f) * wV[row * DX + cc];
  }
  if (tid < 16) sxm[tid] = mask[mtile * 16 + tid] ? 1.f : 0.f;
  __syncthreads();
  // x_out GEMM
  for (int t = 0; t < 2; ++t) {
    int nt = wave * 2 + t;
    v8f acc = {};
    acc = wmma_k(sT, 260, Wxo + nt * 16, DX, DX, lane, acc);
#pragma unroll
    for (int v = 0; v < 8; ++v) {
      int m = v + 8 * half, cc = nt * 16 + r;
      sX1[m * 260 + cc] = acc[v] + bxo[cc];
    }
  }
  __syncthreads();
  for (int idx = tid; idx < 16 * 256; idx += 256) {
    int rr = idx >> 8, cc = idx & 255;
    sX1[rr * 260 + cc] = sX1[rr * 260 + cc] * sxm[rr] + X[(mtile * 16 + rr) * DX + cc];
  }
  __syncthreads();
  ln_rows16(sX1, 260, DX, gX1, bX1, red, tid);
  // chunked FFN: keep output accumulators in registers across chunks
  v8f ao0 = {}, ao1 = {};
  const int nt0 = wave * 2, nt1 = wave * 2 + 1;
  for (int chunk = 0; chunk < 4; ++chunk) {
    for (int t = 0; t < 2; ++t) {
      int nt = wave * 2 + t;
      v8f acc = {};
      acc = wmma_k(sX1, 260, W1 + chunk * 256 + nt * 16, DFFX, DX, lane, acc);
#pragma unroll
      for (int v = 0; v < 8; ++v) {
        int m = v + 8 * half, cc = nt * 16 + r;
        sH[m * 260 + cc] = fmaxf(acc[v] + b1[chunk * 256 + cc], 0.f);
      }
    }
    __syncthreads();
    ao0 = wmma_k(sH, 260, W2 + (chunk * 256) * DX + nt0 * 16, DX, 256, lane, ao0);
    ao1 = wmma_k(sH, 260, W2 + (chunk * 256) * DX + nt1 * 16, DX, 256, lane, ao1);
    __syncthreads();
  }
#pragma unroll
  for (int v = 0; v < 8; ++v) {
    int m = v + 8 * half;
    int c0 = nt0 * 16 + r, c1 = nt1 * 16 + r;
    sT[m * 260 + c0] = ao0[v] + b2[c0] + sX1[m * 260 + c0];
    sT[m * 260 + c1] = ao1[v] + b2[c1] + sX1[m * 260 + c1];
  }
  __syncthreads();
  ln_rows16(sT, 260, DX, gX2, bX2, red, tid);
  for (int idx = tid; idx < 16 * 256; idx += 256) {
    int rr = idx >> 8, cc = idx & 255;
    outX[(mtile * 16 + rr) * DX + cc] = sT[rr * 260 + cc];
  }
}

// ---- y path (tiny): new_y + residual/LN + FFN + LN ----
__global__ void k_newy(const float* __restrict__ y, const float* __restrict__ sx,
                       const float* __restrict__ se,
                       const float* Wyy, const float* byy,
                       const float* Wxy, const float* bxy,
                       const float* Wey, const float* bey,
                       const float* Wo1, const float* bo1,
                       const float* Wo2, const float* bo2,
                       const float* g1, const float* be1,
                       const float* Wl1, const float* bl1,
                       const float* Wl2, const float* bl2,
                       const float* g2, const float* be2, float* outY) {
  int b = blockIdx.x, tid = threadIdx.x;
  __shared__ float sy[DY], t0[DY], t1v[DY], hid[256], red2[2];
  if (tid < DY) sy[tid] = y[b * DY + tid];
  __syncthreads();
  if (tid < DY) {
    float a = byy[tid] + bxy[tid] + bey[tid];
    for (int k = 0; k < DY; ++k) a += sy[k] * Wyy[k * DY + tid];
    for (int k = 0; k < 1024; ++k) a += sx[b * 1024 + k] * Wxy[k * DY + tid];
    for (int k = 0; k < 256; ++k) a += se[b * 256 + k] * Wey[k * DY + tid];
    t0[tid] = a;
  }
  __syncthreads();
  if (tid < DY) {
    float a = bo1[tid];
    for (int k = 0; k < DY; ++k) a += t0[k] * Wo1[k * DY + tid];
    t1v[tid] = fmaxf(a, 0.f);
  }
  __syncthreads();
  if (tid < DY) {
    float a = bo2[tid];
    for (int k = 0; k < DY; ++k) a += t1v[k] * Wo2[k * DY + tid];
    t0[tid] = sy[tid] + a;
  }
  __syncthreads();
  if (tid == 0) {
    float s = 0.f, s2 = 0.f;
    for (int k = 0; k < DY; ++k) { s += t0[k]; s2 += t0[k] * t0[k]; }
    float mean = s / DY;
    red2[0] = mean; red2[1] = rsqrtf(s2 / DY - mean * mean + LN_EPS);
  }
  __syncthreads();
  if (tid < DY) t0[tid] = (t0[tid] - red2[0]) * red2[1] * g1[tid] + be1[tid];
  __syncthreads();
  {
    float a = bl1[tid];
    for (int k = 0; k < DY; ++k) a += t0[k] * Wl1[k * 256 + tid];
    hid[tid] = fmaxf(a, 0.f);
  }
  __syncthreads();
  if (tid < DY) {
    float a = bl2[tid];
    for (int k = 0; k < 256; ++k) a += hid[k] * Wl2[k * DY + tid];
    t1v[tid] = t0[tid] + a;
  }
  __syncthreads();
  if (tid == 0) {
    float s = 0.f, s2 = 0.f;
    for (int k = 0; k < DY; ++k) { s += t1v[k]; s2 += t1v[k] * t1v[k]; }
    float mean = s / DY;
    red2[0] = mean; red2[1] = rsqrtf(s2 / DY - mean * mean + LN_EPS);
  }
  __syncthreads();
  if (tid < DY) outY[b * DY + tid] = (t1v[tid] - red2[0]) * red2[1] * g2[tid] + be2[tid];
}

extern "C" void kernel_launch(void* const* d_in, const int* in_sizes, int n_in,
                              void* d_out, int out_size, void* d_ws, size_t ws_size,
                              hipStream_t stream) {
  const float* X = (const float*)d_in[0];
  const float* E = (const float*)d_in[1];
  const float* y = (const float*)d_in[2];
  const unsigned char* mask = (const unsigned char*)d_in[3];
  const float* P[60];
  for (int i = 4; i < 60 && i < n_in; ++i) P[i] = (const float*)d_in[i];

  float* ws = (float*)d_ws;
  float* Q  = ws;
  float* K  = ws + 262144;
  float* V  = ws + 524288;
  float* wV = ws + 786432;
  float* yv = ws + 1048576;           // 4 x [ye_add|ye_mul|yx_add|yx_mul]
  float* sx = yv + 4096;              // (4,1024)
  float* se = sx + 4096;              // (4,256)

  float* outX = (float*)d_out;        // (4,256,256)
  float* outE = outX + 262144;        // (4,256,256,64)
  float* outY = outX + 4456448;       // (4,64)

  k_yvec<<<NB, 256, 0, stream>>>(y, P[16], P[17], P[14], P[15],
                                 P[20], P[21], P[18], P[19], yv);
  k_qkv<<<64 * 6, 256, 0, stream>>>(X, mask, P[4], P[5], P[6], P[7], P[8], P[9],
                                    Q, K, V);
  k_statsX<<<NB, 256, 0, stream>>>(X, sx);
  k_statsE<<<NB, 256, 0, stream>>>(E, se);
  k_edge<<<NB * NN, 256, 0, stream>>>(E, mask, Q, K, V, yv,
                                      P[12], P[13],   // e_mul
                                      P[10], P[11],   // e_add
                                      P[30], P[31],   // e_out
                                      P[48], P[49],   // normE1
                                      P[44], P[45],   // linE1
                                      P[46], P[47],   // linE2
                                      P[50], P[51],   // normE2
                                      outE, wV);
  k_xfinal<<<64, 256, 0, stream>>>(X, mask, wV, yv,
                                   P[28], P[29],      // x_out
                                   P[40], P[41],      // normX1
                                   P[36], P[37],      // linX1
                                   P[38], P[39],      // linX2
                                   P[42], P[43],      // normX2
                                   outX);
  k_newy<<<NB, 256, 0, stream>>>(y, sx, se,
                                 P[22], P[23], P[24], P[25], P[26], P[27],
                                 P[32], P[33], P[34], P[35],
                                 P[56], P[57], P[52], P[53], P[54], P[55],
                                 P[58], P[59], outY);
}